// SinkhornDistance_60292750901877
// MI455X (gfx1250) — compile-verified
//
#include <hip/hip_runtime.h>
#include <math.h>

#define B_ 8
#define N_ 2048
#define D_ 512
#define EPS_ 0.1f
#define INV_EPS 10.0f
#define THRESH_ 0.1f
#define MAX_ITER_ 100

typedef __attribute__((ext_vector_type(2))) float v2f;
typedef __attribute__((ext_vector_type(8))) float v8f;

// ---------------- init: u=v=0, done=0, mu=nu=1/N ----------------
__global__ void sk_init_kernel(float* __restrict__ u, float* __restrict__ v,
                               float* __restrict__ mu, float* __restrict__ nu,
                               int* __restrict__ done) {
  int idx = blockIdx.x * blockDim.x + threadIdx.x;
  if (idx < B_ * N_) {
    u[idx] = 0.f; v[idx] = 0.f;
    mu[idx] = 1.0f / N_; nu[idx] = 1.0f / N_;
  }
  if (idx == 0) *done = 0;
}

// ---------------- row inverse norms of x and y ----------------
__global__ void sk_norms_kernel(const float* __restrict__ x, const float* __restrict__ y,
                                float* __restrict__ invnx, float* __restrict__ invny) {
  __shared__ float sm[128];
  int r = blockIdx.x;
  const float* src;
  float* dst;
  if (r < B_ * N_) { src = x + (size_t)r * D_;            dst = invnx + r; }
  else             { src = y + (size_t)(r - B_*N_) * D_;  dst = invny + (r - B_*N_); }
  int t = threadIdx.x;
  float s = 0.f;
  #pragma unroll
  for (int q = 0; q < 4; ++q) { float a = src[t + q * 128]; s += a * a; }
  sm[t] = s; __syncthreads();
  for (int off = 64; off > 0; off >>= 1) { if (t < off) sm[t] += sm[t + off]; __syncthreads(); }
  if (t == 0) *dst = rsqrtf(sm[0]);
}

// ---------------- C = 1 - xn @ yn^T via V_WMMA_F32_16X16X4_F32 ----------------
// One wave computes a 64x64 tile as a 4x4 grid of 16x16 WMMA tiles.
// Normalization is bilinear, so inv-norm scales are applied once at write-out.
__global__ __launch_bounds__(256, 1)
void sk_gemm_cost_kernel(const float* __restrict__ x, const float* __restrict__ y,
                         const float* __restrict__ invnx, const float* __restrict__ invny,
                         float* __restrict__ Cout) {
  int gtid = blockIdx.x * blockDim.x + threadIdx.x;
  int wave = gtid >> 5;
  int lane = threadIdx.x & 31;

  int b = wave >> 10;            // 32x32 = 1024 wave-tiles per batch
  int t = wave & 1023;
  int ti = t >> 5, tj = t & 31;
  int r0 = ti * 64, c0 = tj * 64;

  const float* xb = x + (size_t)b * N_ * D_;
  const float* yb = y + (size_t)b * N_ * D_;
  const float* inxb = invnx + b * N_;
  const float* inyb = invny + b * N_;

  int lrow  = lane & 15;         // row/col within a 16-tile
  int khalf = (lane >> 4) << 1;  // 0 or 2
  int ghalf = (lane >> 4) << 3;  // 0 or 8 (C/D row offset)

  v8f acc[4][4];
  v8f zero = {0.f,0.f,0.f,0.f,0.f,0.f,0.f,0.f};
  #pragma unroll
  for (int m = 0; m < 4; ++m)
    #pragma unroll
    for (int n = 0; n < 4; ++n) acc[m][n] = zero;

  const float* px[4];
  const float* py[4];
  #pragma unroll
  for (int m = 0; m < 4; ++m) px[m] = xb + (size_t)(r0 + m * 16 + lrow) * D_ + khalf;
  #pragma unroll
  for (int n = 0; n < 4; ++n) py[n] = yb + (size_t)(c0 + n * 16 + lrow) * D_ + khalf;

  #pragma unroll 2
  for (int k0 = 0; k0 < D_; k0 += 4) {
    v2f afr[4], bfr[4];
    #pragma unroll
    for (int m = 0; m < 4; ++m) afr[m] = *(const v2f*)(px[m] + k0);
    #pragma unroll
    for (int n = 0; n < 4; ++n) bfr[n] = *(const v2f*)(py[n] + k0);
    #pragma unroll
    for (int m = 0; m < 4; ++m)
      #pragma unroll
      for (int n = 0; n < 4; ++n)
        acc[m][n] = __builtin_amdgcn_wmma_f32_16x16x4_f32(
            false, afr[m], false, bfr[n], (short)0, acc[m][n], false, false);
  }

  float* Cb = Cout + (size_t)b * N_ * N_;
  #pragma unroll
  for (int m = 0; m < 4; ++m) {
    float srow[8];
    #pragma unroll
    for (int g = 0; g < 8; ++g) srow[g] = inxb[r0 + m * 16 + ghalf + g];
    #pragma unroll
    for (int n = 0; n < 4; ++n) {
      int col = c0 + n * 16 + lrow;
      float scol = inyb[col];
      #pragma unroll
      for (int g = 0; g < 8; ++g) {
        int row = r0 + m * 16 + ghalf + g;
        Cb[(size_t)row * N_ + col] = 1.0f - acc[m][n][g] * (srow[g] * scol);
      }
    }
  }
}

// ---------------- u update: u_new = eps*(log_mu - lse_j((v_j - C_ij)/eps)) ----------------
__global__ void sk_row_update_kernel(const float* __restrict__ C, float* __restrict__ u,
                                     const float* __restrict__ v, float* __restrict__ du,
                                     const int* __restrict__ done) {
  if (*done) return;
  __shared__ float sm[256];
  int bi = blockIdx.x;                 // bi = b*N + i
  int b = bi >> 11;
  const float* Crow = C + (size_t)bi * N_;
  const float* vb = v + b * N_;
  int t = threadIdx.x;
  float vals[8];
  float m = -INFINITY;
  #pragma unroll
  for (int q = 0; q < 8; ++q) {
    int j = t + q * 256;
    float val = (vb[j] - Crow[j]) * INV_EPS;
    vals[q] = val;
    m = fmaxf(m, val);
  }
  sm[t] = m; __syncthreads();
  for (int off = 128; off > 0; off >>= 1) { if (t < off) sm[t] = fmaxf(sm[t], sm[t + off]); __syncthreads(); }
  m = sm[0]; __syncthreads();
  float s = 0.f;
  #pragma unroll
  for (int q = 0; q < 8; ++q) s += __expf(vals[q] - m);
  sm[t] = s; __syncthreads();
  for (int off = 128; off > 0; off >>= 1) { if (t < off) sm[t] += sm[t + off]; __syncthreads(); }
  if (t == 0) {
    float lse = m + __logf(sm[0]);
    const float log_mu = __logf(1.0f / N_ + 1e-8f);
    float unew = EPS_ * (log_mu - lse);
    float uold = u[bi];
    u[bi] = unew;
    du[bi] = fabsf(unew - uold);
  }
}

// ---------------- column logsumexp partials (8 segments of 256 rows each) ----------------
__global__ void sk_col_part_kernel(const float* __restrict__ C, const float* __restrict__ u,
                                   float* __restrict__ partm, float* __restrict__ parts,
                                   const int* __restrict__ done) {
  if (*done) return;
  int g = blockIdx.x * blockDim.x + threadIdx.x;  // [0, B*8*N)
  int j = g & (N_ - 1);
  int seg = (g >> 11) & 7;
  int b = g >> 14;
  const float* Cb = C + (size_t)b * N_ * N_;
  const float* ub = u + b * N_;
  float m = -INFINITY, s = 0.f;
  int i0 = seg * 256;
  for (int q = 0; q < 256; ++q) {
    int i = i0 + q;
    float val = (ub[i] - Cb[(size_t)i * N_ + j]) * INV_EPS;
    if (val <= m) { s += __expf(val - m); }
    else { s = s * __expf(m - val) + 1.0f; m = val; }
  }
  partm[g] = m;
  parts[g] = s;
}

// ---------------- combine partials: v_new = eps*(log_nu - lse_i((u_i - C_ij)/eps)) ----------------
__global__ void sk_col_combine_kernel(float* __restrict__ v, const float* __restrict__ partm,
                                      const float* __restrict__ parts, const int* __restrict__ done) {
  if (*done) return;
  int idx = blockIdx.x * blockDim.x + threadIdx.x;  // [0, B*N)
  int b = idx >> 11, j = idx & (N_ - 1);
  float m = -INFINITY;
  #pragma unroll
  for (int seg = 0; seg < 8; ++seg) m = fmaxf(m, partm[(b * 8 + seg) * N_ + j]);
  float s = 0.f;
  #pragma unroll
  for (int seg = 0; seg < 8; ++seg)
    s += parts[(b * 8 + seg) * N_ + j] * __expf(partm[(b * 8 + seg) * N_ + j] - m);
  float lse = m + __logf(s);
  const float log_nu = __logf(1.0f / N_ + 1e-8f);
  v[idx] = EPS_ * (log_nu - lse);
}

// ---------------- deterministic err reduction + convergence flag ----------------
__global__ void sk_finalize_kernel(const float* __restrict__ du, int* __restrict__ done) {
  if (*done) return;
  __shared__ float sm[256];
  int t = threadIdx.x;
  float s = 0.f;
  for (int q = 0; q < (B_ * N_) / 256; ++q) s += du[t + q * 256];
  sm[t] = s; __syncthreads();
  for (int off = 128; off > 0; off >>= 1) { if (t < off) sm[t] += sm[t + off]; __syncthreads(); }
  if (t == 0) {
    float err = sm[0] / (float)B_;   // sum over N, mean over batch
    if (err < THRESH_) *done = 1;
  }
}

// ---------------- pi = exp((u_i + v_j - C_ij)/eps), per-row sums of pi*C ----------------
__global__ void sk_pi_kernel(const float* __restrict__ C, const float* __restrict__ u,
                             const float* __restrict__ v, float* __restrict__ pi,
                             float* __restrict__ rowsum) {
  __shared__ float sm[256];
  int bi = blockIdx.x;
  int b = bi >> 11;
  const float* Crow = C + (size_t)bi * N_;
  float* prow = pi + (size_t)bi * N_;
  const float* vb = v + b * N_;
  float ui = u[bi];
  int t = threadIdx.x;
  float s = 0.f;
  #pragma unroll
  for (int q = 0; q < 8; ++q) {
    int j = t + q * 256;
    float c = Crow[j];
    float p = __expf((ui + vb[j] - c) * INV_EPS);
    prow[j] = p;
    s += p * c;
  }
  sm[t] = s; __syncthreads();
  for (int off = 128; off > 0; off >>= 1) { if (t < off) sm[t] += sm[t + off]; __syncthreads(); }
  if (t == 0) rowsum[bi] = sm[0];
}

// ---------------- per-batch cost (deterministic) ----------------
__global__ void sk_cost_kernel(const float* __restrict__ rowsum, float* __restrict__ cost) {
  __shared__ float sm[256];
  int b = blockIdx.x, t = threadIdx.x;
  float s = 0.f;
  for (int q = 0; q < N_ / 256; ++q) s += rowsum[b * N_ + t + q * 256];
  sm[t] = s; __syncthreads();
  for (int off = 128; off > 0; off >>= 1) { if (t < off) sm[t] += sm[t + off]; __syncthreads(); }
  if (t == 0) cost[b] = sm[0];
}

extern "C" void kernel_launch(void* const* d_in, const int* in_sizes, int n_in,
                              void* d_out, int out_size, void* d_ws, size_t ws_size,
                              hipStream_t stream) {
  const float* x = (const float*)d_in[0];
  const float* y = (const float*)d_in[1];

  float* out  = (float*)d_out;
  float* cost = out;                       // [8]
  float* mu   = out + 8;                   // [8*2048]
  float* nu   = mu + B_ * N_;              // [8*2048]
  float* pi   = nu + B_ * N_;              // [8*2048*2048]
  float* C    = pi + (size_t)B_ * N_ * N_; // [8*2048*2048]

  float* ws     = (float*)d_ws;
  float* invnx  = ws;                        // 16384
  float* invny  = invnx + B_ * N_;           // 16384
  float* u      = invny + B_ * N_;           // 16384
  float* v      = u + B_ * N_;               // 16384
  float* partm  = v + B_ * N_;               // 131072
  float* parts  = partm + B_ * 8 * N_;       // 131072
  float* du     = parts + B_ * 8 * N_;       // 16384
  float* rowsum = du + B_ * N_;              // 16384
  int*   done   = (int*)(rowsum + B_ * N_);  // 1

  sk_init_kernel<<<(B_ * N_ + 255) / 256, 256, 0, stream>>>(u, v, mu, nu, done);
  sk_norms_kernel<<<2 * B_ * N_, 128, 0, stream>>>(x, y, invnx, invny);
  sk_gemm_cost_kernel<<<(B_ * 1024) / 8, 256, 0, stream>>>(x, y, invnx, invny, C);

  for (int it = 0; it < MAX_ITER_; ++it) {
    sk_row_update_kernel<<<B_ * N_, 256, 0, stream>>>(C, u, v, du, done);
    sk_col_part_kernel<<<(B_ * 8 * N_) / 256, 256, 0, stream>>>(C, u, partm, parts, done);
    sk_col_combine_kernel<<<(B_ * N_) / 256, 256, 0, stream>>>(v, partm, parts, done);
    sk_finalize_kernel<<<1, 256, 0, stream>>>(du, done);
  }

  sk_pi_kernel<<<B_ * N_, 256, 0, stream>>>(C, u, v, pi, rowsum);
  sk_cost_kernel<<<B_, 256, 0, stream>>>(rowsum, cost);
}